// CustomDFAModel_283467842494
// MI455X (gfx1250) — compile-verified
//
#include <hip/hip_runtime.h>

// ---------------------------------------------------------------------------
// Multi-scale deformable attention aggregation for MI455X (gfx1250).
// Gather-bound: value tensor (184MB) fits in 192MB L2; optimize for coalesced
// row-segment gathers + memory-level parallelism. Attention weights staged to
// LDS via the Tensor Data Mover (async-tensor path) when available.
// ---------------------------------------------------------------------------

#define BS   2
#define AQ   900
#define PP   13
#define NC   6
#define NL   4
#define NG   8
#define CCH  256
#define NS   (PP * NC * NL)   /* 312 samples per query */

#ifndef __has_builtin
#define __has_builtin(x) 0
#endif

#if defined(__gfx1250__) && __has_builtin(__builtin_amdgcn_tensor_load_to_lds) && \
    __has_builtin(__builtin_amdgcn_s_wait_tensorcnt)
#define USE_TDM 1
#else
#define USE_TDM 0
#endif

typedef unsigned int uv4 __attribute__((ext_vector_type(4)));
typedef int          iv8 __attribute__((ext_vector_type(8)));
typedef int          iv4 __attribute__((ext_vector_type(4)));

__global__ __launch_bounds__(256)
void msda_gfx1250_kernel(const float* __restrict__ value,
                         const int*   __restrict__ shapes,
                         const int*   __restrict__ starts,
                         const float* __restrict__ locs,
                         const float* __restrict__ attw,
                         float*       __restrict__ out,
                         int K)
{
    __shared__ int   sIdx[NS][4];        // corner row indices
    __shared__ float sW[NS][4];          // bilinear corner weights
    __shared__ float sAtt[NS * NG];      // attention weights (2496 f32)
    __shared__ float sRed[4][CCH];       // cross-set reduction buffer

    const int tid = threadIdx.x;
    const int ba  = blockIdx.x;          // 0 .. BS*AQ-1
    const int b   = ba / AQ;

    const float* attw_g = attw + (size_t)ba * (NS * NG);

#if USE_TDM
    // --- Stage attention weights to LDS with the Tensor Data Mover (wave 0) ---
    if (tid < 32) {
        unsigned long long ga  = (unsigned long long)(size_t)attw_g;
        unsigned           lds = (unsigned)(size_t)(&sAtt[0]); // flat low 32 = LDS byte addr
        uv4 g0 = { 1u,                                          // count=1, user mode
                   lds,                                         // lds_addr
                   (unsigned)(ga & 0xffffffffu),                // global_addr[31:0]
                   (unsigned)((ga >> 32) & 0x01ffffffu)         // global_addr[56:32]
                       | 0x80000000u };                         // type = 2 ("image")
        iv8 g1 = { (int)(2u << 16),                             // data_size = 4B
                   (int)(((NS * NG) & 0xffff) << 16),           // tensor_dim0 lo16 = 2496
                   (int)(1 << 16),                              // dim0 hi=0 | tensor_dim1 lo16 = 1
                   (int)((NS * NG) << 16),                      // dim1 hi=0 | tile_dim0 = 2496
                   1,                                           // tile_dim1 = 1, tile_dim2 = 0
                   NS * NG,                                     // tensor_dim0_stride lo32
                   0, 0 };
        iv4 z4 = { 0, 0, 0, 0 };
        iv8 z8 = { 0, 0, 0, 0, 0, 0, 0, 0 };
        // 6-arg (clang-23 / therock-10.0) form:
        // (g0:u32x4, g1:i32x8, i32x4, i32x4, i32x8, cpol)
        __builtin_amdgcn_tensor_load_to_lds(g0, g1, z4, z4, z8, 0);
    }
#else
    for (int i = tid; i < NS * NG; i += 256) sAtt[i] = attw_g[i];
#endif

    // --- Phase 1: per-sample bilinear metadata (overlaps TDM transfer) ---
    for (int s = tid; s < NS; s += 256) {
        const int p   = s / (NC * NL);
        const int r   = s - p * (NC * NL);
        const int cam = r >> 2;
        const int lvl = r & 3;
        const int sh  = cam * NL + lvl;
        const int H   = shapes[sh * 2 + 0];
        const int W   = shapes[sh * 2 + 1];
        const int st  = starts[sh];
        const float lx = locs[(((size_t)ba * PP + p) * NC + cam) * 2 + 0];
        const float ly = locs[(((size_t)ba * PP + p) * NC + cam) * 2 + 1];
        const float x  = lx * (float)W - 0.5f;
        const float y  = ly * (float)H - 0.5f;
        const float xf = floorf(x), yf = floorf(y);
        const float dx = x - xf,    dy = y - yf;
        const int   x0 = (int)xf,   y0 = (int)yf;
#pragma unroll
        for (int c = 0; c < 4; ++c) {
            const int oy = c >> 1, ox = c & 1;
            const int yi = y0 + oy, xi = x0 + ox;
            float w = (oy ? dy : 1.0f - dy) * (ox ? dx : 1.0f - dx);
            if (yi < 0 || yi >= H || xi < 0 || xi >= W) w = 0.0f;
            const int yc = min(max(yi, 0), H - 1);
            const int xc = min(max(xi, 0), W - 1);
            sIdx[s][c] = st + yc * W + xc;
            sW[s][c]   = w;
        }
    }

#if USE_TDM
    if (tid < 32) __builtin_amdgcn_s_wait_tensorcnt((short)0);
#endif
    __syncthreads();

    // --- Phase 2: gather + weighted accumulation ---
    // 64 channel-quads (float4) x 4 sample-sets. Each wave: 32 lanes x 16B =
    // 512B contiguous per corner row -> fully coalesced global_load_b128.
    const int q   = tid & 63;        // channel quad: channels 4q..4q+3
    const int set = tid >> 6;        // sample phase 0..3
    const int grp = q >> 3;          // attention group = (4q)/32

    const float4* __restrict__ vb =
        (const float4*)(value + (size_t)b * (size_t)K * CCH);

    float4 acc = { 0.f, 0.f, 0.f, 0.f };
#pragma unroll 2
    for (int s = set; s < NS; s += 4) {
        const int   i0 = sIdx[s][0], i1 = sIdx[s][1], i2 = sIdx[s][2], i3 = sIdx[s][3];
        const float w0 = sW[s][0],   w1 = sW[s][1],   w2 = sW[s][2],   w3 = sW[s][3];
        const float aw = sAtt[s * NG + grp];

        const float4 v0 = vb[(size_t)i0 * 64 + q];
        const float4 v1 = vb[(size_t)i1 * 64 + q];
        const float4 v2 = vb[(size_t)i2 * 64 + q];
        const float4 v3 = vb[(size_t)i3 * 64 + q];

        acc.x += aw * (w0 * v0.x + w1 * v1.x + w2 * v2.x + w3 * v3.x);
        acc.y += aw * (w0 * v0.y + w1 * v1.y + w2 * v2.y + w3 * v3.y);
        acc.z += aw * (w0 * v0.z + w1 * v1.z + w2 * v2.z + w3 * v3.z);
        acc.w += aw * (w0 * v0.w + w1 * v1.w + w2 * v2.w + w3 * v3.w);

        const int sn = s + 4;
        if (sn < NS) {   // warm next sample's corner rows (L2 -> near cache)
            __builtin_prefetch(&vb[(size_t)sIdx[sn][0] * 64 + q], 0, 3);
            __builtin_prefetch(&vb[(size_t)sIdx[sn][3] * 64 + q], 0, 3);
        }
    }

    ((float4*)sRed[set])[q] = acc;
    __syncthreads();

    if (tid < 64) {
        const float4 r0 = ((const float4*)sRed[0])[tid];
        const float4 r1 = ((const float4*)sRed[1])[tid];
        const float4 r2 = ((const float4*)sRed[2])[tid];
        const float4 r3 = ((const float4*)sRed[3])[tid];
        float4 o;
        o.x = r0.x + r1.x + r2.x + r3.x;
        o.y = r0.y + r1.y + r2.y + r3.y;
        o.z = r0.z + r1.z + r2.z + r3.z;
        o.w = r0.w + r1.w + r2.w + r3.w;
        ((float4*)(out + (size_t)ba * CCH))[tid] = o;
    }
}

extern "C" void kernel_launch(void* const* d_in, const int* in_sizes, int n_in,
                              void* d_out, int out_size, void* d_ws, size_t ws_size,
                              hipStream_t stream) {
    const float* value  = (const float*)d_in[0];
    const int*   shapes = (const int*)d_in[1];
    const int*   starts = (const int*)d_in[2];
    const float* locs   = (const float*)d_in[3];
    const float* attw   = (const float*)d_in[4];
    float*       out    = (float*)d_out;

    const int K = in_sizes[0] / (BS * CCH);   // 89760

    dim3 grid(BS * AQ);   // 1800 query blocks
    dim3 block(256);      // 8 wave32s
    hipLaunchKernelGGL(msda_gfx1250_kernel, grid, block, 0, stream,
                       value, shapes, starts, locs, attw, out, K);
}